// StochasticSage_1692217114896
// MI455X (gfx1250) — compile-verified
//
#include <hip/hip_runtime.h>
#include <hip/hip_bf16.h>

// CDNA5 / gfx1250 GraphSAGE forward.
//  - scatter: coalesced float4 gather + global_atomic_add_f32 into agg/deg,
//             global_prefetch_b8 on a future edge's source row.
//  - gemm:    fp32 WMMA (v_wmma_f32_16x16x4_f32). A panels staged in LDS
//             (shared by 8 waves), weights pre-transposed so B fragments are
//             contiguous b64 loads; 2 output tiles per wave for reuse.

typedef __attribute__((ext_vector_type(2))) float v2f;
typedef __attribute__((ext_vector_type(8))) float v8f;

#define DD 256

__global__ __launch_bounds__(256) void zero_f32(float* __restrict__ p, int n) {
    int i = blockIdx.x * 256 + threadIdx.x;
    if (i < n) p[i] = 0.0f;
}

// Wt[c][k] = W[k][c], 256x256, 16x16 tile transpose via LDS.
__global__ __launch_bounds__(256) void transpose256(const float* __restrict__ W,
                                                    float* __restrict__ Wt) {
    __shared__ float tile[16][17];
    const int tx = threadIdx.x & 15;
    const int ty = threadIdx.x >> 4;
    tile[ty][tx] = W[(size_t)(blockIdx.y * 16 + ty) * DD + blockIdx.x * 16 + tx];
    __syncthreads();
    Wt[(size_t)(blockIdx.x * 16 + ty) * DD + blockIdx.y * 16 + tx] = tile[tx][ty];
}

// 4 edges per 256-thread block; 64 lanes per edge, float4 each (=256 floats).
__global__ __launch_bounds__(256) void sage_scatter(
    const float* __restrict__ H, const int* __restrict__ src,
    const int* __restrict__ dst, float* __restrict__ agg,
    float* __restrict__ deg, int nE)
{
    const int t = threadIdx.x;
    const int eLocal = t >> 6;
    const int lane = t & 63;
    const int e = blockIdx.x * 4 + eLocal;
    if (e >= nE) return;
    const int s = src[e];
    const int d = dst[e];
    const float4* hrow = (const float4*)(H + (size_t)s * DD);
    float4 v = hrow[lane];
    float* arow = agg + (size_t)d * DD + lane * 4;
    atomicAdd(arow + 0, v.x);
    atomicAdd(arow + 1, v.y);
    atomicAdd(arow + 2, v.z);
    atomicAdd(arow + 3, v.w);
    if (lane == 0) atomicAdd(deg + d, 1.0f);
    // CDNA5 prefetch (global_prefetch_b8) of a future edge's source row.
    const int e2 = e + 4;
    if (e2 < nE) {
        const float* nxt = H + (size_t)src[e2] * DD + lane * 4;
        __builtin_prefetch(nxt, 0, 1);
    }
}

// out[nd x 256] = relu( Hd @ Ws + (agg * 1/max(deg,1)) @ Wn + bias )
// Block = 8 waves; block owns rows [row0,row0+16) x all 256 cols.
// Wave w owns cols w*16..+16 and w*16+128..+16 (2 tiles, A reuse).
// Wst/Wnt are TRANSPOSED weights: Wt[c][k].
__global__ __launch_bounds__(256) void sage_gemm(
    const float* __restrict__ Hd,   // >= nd x 256, row-major
    const float* __restrict__ Agg,  // nd x 256
    const float* __restrict__ Deg,  // nd
    const float* __restrict__ Wst,  // 256 x 256 (transposed self weight)
    const float* __restrict__ Wnt,  // 256 x 256 (transposed neigh weight)
    const float* __restrict__ Bias, // 256
    float* __restrict__ Out,        // nd x 256
    int nd)
{
    __shared__ float sA[16 * DD];   // self-feature panel
    __shared__ float sM[16 * DD];   // mean (agg*invdeg) panel
    __shared__ float sInv[16];

    const int t = threadIdx.x;
    const int row0 = blockIdx.x * 16;

    if (t < 16) {
        int r = row0 + t;
        int rA = (r < nd) ? r : (nd - 1);
        sInv[t] = 1.0f / fmaxf(Deg[rA], 1.0f);
    }
    __syncthreads();

    // Cooperative panel load: 2 x (16 rows x 64 float4), 4 float4 per thread.
    {
        const float4* HdV = (const float4*)Hd;
        const float4* AgV = (const float4*)Agg;
        float4* sAv = (float4*)sA;
        float4* sMv = (float4*)sM;
        #pragma unroll
        for (int j = 0; j < 4; ++j) {
            int idx = t + j * 256;          // 0..1023
            int rl = idx >> 6;              // local row 0..15
            int c4 = idx & 63;              // float4 column
            int r = row0 + rl;
            int rA = (r < nd) ? r : (nd - 1);
            float4 hv = HdV[(size_t)rA * 64 + c4];
            float4 av = AgV[(size_t)rA * 64 + c4];
            float inv = sInv[rl];
            av.x *= inv; av.y *= inv; av.z *= inv; av.w *= inv;
            sAv[idx] = hv;
            sMv[idx] = av;
        }
    }
    __syncthreads();

    const int lane = t & 31;
    const int wave = t >> 5;                 // 0..7
    const int half = lane >> 4;              // K-pair select
    const int l16  = lane & 15;              // A row / B,D col within tile
    const int col0 = wave * 16;              // tile 0 cols
    const int col1 = col0 + 128;             // tile 1 cols

    const float* sArow = sA + l16 * DD;
    const float* sMrow = sM + l16 * DD;
    const float* bs0 = Wst + (size_t)(col0 + l16) * DD;
    const float* bs1 = Wst + (size_t)(col1 + l16) * DD;
    const float* bn0 = Wnt + (size_t)(col0 + l16) * DD;
    const float* bn1 = Wnt + (size_t)(col1 + l16) * DD;

    v8f acc0 = {};
    v8f acc1 = {};
    for (int k = 0; k < DD; k += 4) {
        const int kk = k + 2 * half;
        v2f a;  a.x  = sArow[kk]; a.y  = sArow[kk + 1];   // ds_load_b64
        v2f am; am.x = sMrow[kk]; am.y = sMrow[kk + 1];
        v2f b0; b0.x = bs0[kk];   b0.y = bs0[kk + 1];     // global b64 (L2-hot)
        v2f b1; b1.x = bs1[kk];   b1.y = bs1[kk + 1];
        v2f c0; c0.x = bn0[kk];   c0.y = bn0[kk + 1];
        v2f c1; c1.x = bn1[kk];   c1.y = bn1[kk + 1];
        acc0 = __builtin_amdgcn_wmma_f32_16x16x4_f32(false, a,  false, b0, (short)0, acc0, false, false);
        acc1 = __builtin_amdgcn_wmma_f32_16x16x4_f32(false, a,  false, b1, (short)0, acc1, false, false);
        acc0 = __builtin_amdgcn_wmma_f32_16x16x4_f32(false, am, false, c0, (short)0, acc0, false, false);
        acc1 = __builtin_amdgcn_wmma_f32_16x16x4_f32(false, am, false, c1, (short)0, acc1, false, false);
    }

    // D layout: lane L, vgpr v -> row v + 8*(L/16), col L%16.
    const float bias0 = Bias[col0 + l16];
    const float bias1 = Bias[col1 + l16];
    #pragma unroll
    for (int v = 0; v < 8; ++v) {
        int rr = row0 + v + 8 * half;
        if (rr < nd) {
            float v0 = fmaxf(acc0[v] + bias0, 0.0f);      // relu all 3 layers
            float v1 = fmaxf(acc1[v] + bias1, 0.0f);
            Out[(size_t)rr * DD + col0 + l16] = v0;
            Out[(size_t)rr * DD + col1 + l16] = v1;
        }
    }
}

extern "C" void kernel_launch(void* const* d_in, const int* in_sizes, int n_in,
                              void* d_out, int out_size, void* d_ws, size_t ws_size,
                              hipStream_t stream) {
    const float* x   = (const float*)d_in[0];
    const float* Wsf[3] = { (const float*)d_in[1], (const float*)d_in[4], (const float*)d_in[7] };
    const float* Wng[3] = { (const float*)d_in[2], (const float*)d_in[5], (const float*)d_in[8] };
    const float* bia[3] = { (const float*)d_in[3], (const float*)d_in[6], (const float*)d_in[9] };
    const int* srcs[3] = { (const int*)d_in[10], (const int*)d_in[12], (const int*)d_in[14] };
    const int* dsts[3] = { (const int*)d_in[11], (const int*)d_in[13], (const int*)d_in[15] };
    const int E[3]  = { in_sizes[10], in_sizes[12], in_sizes[14] };
    const int ND[3] = { 50000, 12000, 4000 };   // NODE_COUNTS[1..3] (static in reference)

    float* ws  = (float*)d_ws;
    float* h1  = ws;                                 // 50000*256
    float* h2  = h1 + (size_t)ND[0] * DD;            // 12000*256
    float* agg = h2 + (size_t)ND[1] * DD;            // 50000*256 (reused per layer)
    float* deg = agg + (size_t)ND[0] * DD;           // 50000
    float* wtb = deg + ND[0];                        // 6 * 256*256 transposed weights

    float* Wst[3], * Wnt[3];
    for (int l = 0; l < 3; ++l) {
        Wst[l] = wtb + (size_t)(2 * l) * DD * DD;
        Wnt[l] = wtb + (size_t)(2 * l + 1) * DD * DD;
    }

    // Pre-transpose all six 256x256 weights (1.5 MB total, L2-resident).
    dim3 tg(16, 16);
    for (int l = 0; l < 3; ++l) {
        transpose256<<<tg, 256, 0, stream>>>(Wsf[l], Wst[l]);
        transpose256<<<tg, 256, 0, stream>>>(Wng[l], Wnt[l]);
    }

    const float* Hin[3] = { x, h1, h2 };
    float* Hout[3] = { h1, h2, (float*)d_out };

    for (int l = 0; l < 3; ++l) {
        const int nd = ND[l];
        const int nAgg = nd * DD;
        zero_f32<<<(nAgg + 255) / 256, 256, 0, stream>>>(agg, nAgg);
        zero_f32<<<(nd + 255) / 256, 256, 0, stream>>>(deg, nd);
        sage_scatter<<<(E[l] + 3) / 4, 256, 0, stream>>>(
            Hin[l], srcs[l], dsts[l], agg, deg, E[l]);
        sage_gemm<<<(nd + 15) / 16, 256, 0, stream>>>(
            Hin[l], agg, deg, Wst[l], Wnt[l], bia[l], Hout[l], nd);
    }
}